// RoPE_51943334478446
// MI455X (gfx1250) — compile-verified
//
#include <hip/hip_runtime.h>

// RoPE: out[b,h,s,i] = sum_j rope[pos[s],i,j] * x[b,h,s,j]
// rope is block-diagonal 2x2 rotations -> elementwise pair rotation.
// B=4, H=16 -> BH=64; S=4096; D=128; rope table [4096,128,128] f32.
//
// Bandwidth-bound: 128MB x-read + 128MB out-write + ~64MB diagonal gather
// => ~14us at 23.3 TB/s. CDNA5 path used: GLOBAL_LOAD_ASYNC_TO_LDS_B32
// scatter-gather of the rotation diagonals into LDS (reused 64x per block),
// s_wait_asynccnt, and NT temporal hints on the zero-reuse x/out streams.

#define BH_TOTAL 64
#define SEQ      4096
#define DIM      128
#define TILE_S   4
#define THREADS  256

typedef float f4 __attribute__((ext_vector_type(4)));

__global__ __launch_bounds__(THREADS) void rope_apply_kernel(
    const float* __restrict__ x,
    const int*   __restrict__ tok,
    const float* __restrict__ rope,
    float*       __restrict__ out)
{
    // [s_local*128 + j]: even j holds cos_k, odd j holds sin_k (k = j/2)
    __shared__ float lds_cs[TILE_S * DIM];

    const int t  = threadIdx.x;
    const int s0 = blockIdx.x * TILE_S;

    // ---- async gather of the rotation diagonals into LDS ----
    // cos_k = rope[p, 2k, 2k]     -> elem off = p*16384 + j*129        (j=2k)
    // sin_k = rope[p, 2k+1, 2k]   -> elem off = p*16384 + j*129 - 1    (j=2k+1)
#pragma unroll
    for (int i = 0; i < (TILE_S * DIM) / THREADS; ++i) {
        const int idx     = t + i * THREADS;      // 0..511
        const int s_local = idx >> 7;             // 0..TILE_S-1
        const int j       = idx & (DIM - 1);      // 0..127
        const int p       = tok[s0 + s_local];
        const unsigned goff = (unsigned)((p * (DIM * DIM) + j * (DIM + 1) - (j & 1)) * 4);
        // low 32 bits of the flat pointer to a __shared__ object == LDS byte offset
        const unsigned loff = (unsigned)(size_t)(const void*)&lds_cs[idx];
        asm volatile("global_load_async_to_lds_b32 %0, %1, %2 offset:0"
                     :: "v"(loff), "v"(goff), "s"(rope)
                     : "memory");
    }
    __builtin_amdgcn_s_wait_asynccnt(0);   // this wave's async loads into LDS done
    __syncthreads();                       // visible to all waves in the block

    // ---- stream all 64 (b,h) rows of this s-tile, float4 per thread ----
    const f4* __restrict__ x4  = (const f4*)x;
    f4*       __restrict__ o4  = (f4*)out;
    const f4*             lds4 = (const f4*)lds_cs;

    const int bh_sub = t >> 7;     // 0..1 : two bh rows per iteration
    const int q      = t & 127;    // float4 index within the 4x128 element tile

    // (cos_k0, sin_k0, cos_k1, sin_k1) for this thread's two pairs
    const f4 c = lds4[q];

    const size_t row4 = (size_t)SEQ * (DIM / 4);                 // f4 stride per bh
    size_t base = ((size_t)bh_sub * SEQ + (size_t)s0) * (DIM / 4) + (size_t)q;

#pragma unroll 4
    for (int it = 0; it < BH_TOTAL / 2; ++it) {
        const f4 v = __builtin_nontemporal_load(&x4[base]);   // TH=NT: zero reuse
        f4 r;
        r.x = c.x * v.x - c.y * v.y;   // cos*xe - sin*xo
        r.y = c.y * v.x + c.x * v.y;   // sin*xe + cos*xo
        r.z = c.z * v.z - c.w * v.w;
        r.w = c.w * v.z + c.z * v.w;
        __builtin_nontemporal_store(r, &o4[base]);            // TH=NT stream-out
        base += 2 * row4;
    }
}

extern "C" void kernel_launch(void* const* d_in, const int* in_sizes, int n_in,
                              void* d_out, int out_size, void* d_ws, size_t ws_size,
                              hipStream_t stream) {
    (void)in_sizes; (void)n_in; (void)d_ws; (void)ws_size; (void)out_size;
    const float* x    = (const float*)d_in[0];
    const int*   tok  = (const int*)d_in[1];
    const float* rope = (const float*)d_in[2];
    float*       out  = (float*)d_out;

    rope_apply_kernel<<<dim3(SEQ / TILE_S), dim3(THREADS), 0, stream>>>(x, tok, rope, out);
}